// retina_polar_learnw_teacher_89017492177064
// MI455X (gfx1250) — compile-verified
//
#include <hip/hip_runtime.h>

// Problem constants (from reference)
#define B        64
#define CCH      3
#define HIN      224
#define WIN      224
#define NPIX     (HIN * WIN)        // 50176 floats = 200704 B
#define CHUNKS   (NPIX / 4)         // 12544 x 16B chunks
#define HG       320                // polar rows (radius)
#define HOUT     32
#define WOUT     64
#define UP       10

// CDNA5 async global->LDS path (probe-verified in round 2: signature is
// (v4i addrspace(1)*, v4i addrspace(3)*, imm offset, imm cpol)).
#if defined(__HIP_DEVICE_COMPILE__) && defined(__gfx1250__) && \
    __has_builtin(__builtin_amdgcn_global_load_async_to_lds_b128) && \
    __has_builtin(__builtin_amdgcn_s_wait_asynccnt)
#define RETINA_ASYNC 1
#else
#define RETINA_ASYNC 0
#endif

#if RETINA_ASYNC
typedef int retina_v4i __attribute__((ext_vector_type(4)));
typedef __attribute__((address_space(1))) retina_v4i retina_gv4i;
typedef __attribute__((address_space(3))) retina_v4i retina_lv4i;
#endif

__global__ __launch_bounds__(1024, 1)
void retina_polar_kernel(const float* __restrict__ x,
                         const float* __restrict__ lt,
                         const float* __restrict__ w,
                         float* __restrict__ out)
{
    // 200704 B image + 1280 B radius table + 512 B angle tables  (~202.5 KB of 320 KB WGP LDS)
    __shared__ float sImg[NPIX];
    __shared__ float sR[HG];      // sample_r[i] * w0 * 112  (pixel-space radius)
    __shared__ float sSin[64];
    __shared__ float sCos[64];

    const int bc  = blockIdx.x;       // (b, c) pair: 0..191
    const int b   = bc / CCH;
    const int tid = threadIdx.x;

    const float w0 = w[2 * b + 0];
    const float w1 = w[2 * b + 1];
    const float lx = lt[2 * b + 0];
    const float ly = lt[2 * b + 1];

    // Pixel-space center: ix = gx*112 + 111.5 with gx = R*s + lx
    //                  => ix = (R*112)*s + (lx*112 + 111.5)
    const float LX = fmaf(lx, 112.f, 111.5f);
    const float LY = fmaf(ly, 112.f, 111.5f);

    const float* gsrc = x + (size_t)bc * NPIX;

    // ---- Stage the 224x224 channel image into LDS (async, ASYNCcnt-tracked) ----
    for (int idx = tid; idx < CHUNKS; idx += 1024) {
#if RETINA_ASYNC
        __builtin_amdgcn_global_load_async_to_lds_b128(
            (retina_gv4i*)(gsrc + 4 * idx),
            (retina_lv4i*)(&sImg[4 * idx]),
            /*offset=*/0, /*cpol=*/0);
#else
        ((float4*)sImg)[idx] = ((const float4*)gsrc)[idx];
#endif
    }

    // ---- Radius table: sample_r[i] * w0 * 112 (log-spaced 0.05..0.8, 320 pts) ----
    if (tid < HG) {
        const float logmin = -2.99573227355399f;        // ln(0.05)
        const float step   = 2.77258872223978f / 319.f; // ln(16)/319
        sR[tid] = __expf(fmaf((float)tid, step, logmin)) * w0 * 112.f;
    }
    // ---- Rotated angle tables: base_ang has period 64 in j, fold in -w1*2pi ----
    if (tid < 64) {
        const float twopi = 6.283185307179586f;
        float ang = twopi * ((float)tid * (1.0f / 64.0f)) - twopi * w1;
        float s, c;
        __sincosf(ang, &s, &c);
        sSin[tid] = s;
        sCos[tid] = c;
    }

#if RETINA_ASYNC
    __builtin_amdgcn_s_wait_asynccnt(0);
#endif
    __syncthreads();

    // ---- Each thread: 2 output pixels, each the mean of a 10x10 sample block ----
    for (int o = tid; o < HOUT * WOUT; o += 1024) {
        const int oy = o >> 6;       // 0..31
        const int ox = o & 63;       // 0..63

        // Hoist the 10 rotated (sin,cos) pairs: invariant across dy.
        float sA[UP], cA[UP];
#pragma unroll
        for (int dx = 0; dx < UP; ++dx) {
            const int jm = (ox * UP + dx) & 63;   // 640 % 64 periodicity
            sA[dx] = sSin[jm];
            cA[dx] = sCos[jm];
        }

        float acc = 0.f;
        for (int dy = 0; dy < UP; ++dy) {
            const float Rp = sR[oy * UP + dy];    // already in pixel units
#pragma unroll
            for (int dx = 0; dx < UP; ++dx) {
                float ix = fmaf(Rp, sA[dx], LX);
                float iy = fmaf(Rp, cA[dx], LY);
                ix = fminf(fmaxf(ix, 0.f), 223.f);
                iy = fminf(fmaxf(iy, 0.f), 223.f);
                const int   x0 = (int)ix;             // floor: ix >= 0
                const int   y0 = (int)iy;
                const float wx = ix - (float)x0;
                const float wy = iy - (float)y0;
                // Single base pointer + constant offsets -> 2x ds_load_2addr_b32.
                // Border over-reads hit LDS OOB which returns 0; the matching
                // lerp weight is exactly 0 there, so the result is unchanged.
                const float* p = &sImg[y0 * WIN + x0];
                const float v00 = p[0];
                const float v01 = p[1];
                const float v10 = p[WIN];
                const float v11 = p[WIN + 1];
                const float top = fmaf(wx, v01 - v00, v00);
                const float bot = fmaf(wx, v11 - v10, v10);
                acc += fmaf(wy, bot - top, top);
            }
        }
        out[(size_t)bc * (HOUT * WOUT) + o] = acc * 0.01f;
    }
}

extern "C" void kernel_launch(void* const* d_in, const int* in_sizes, int n_in,
                              void* d_out, int out_size, void* d_ws, size_t ws_size,
                              hipStream_t stream) {
    const float* x  = (const float*)d_in[0];  // [64,3,224,224]
    const float* lt = (const float*)d_in[1];  // [64,2]
    // d_in[2] = s_t is unused by the reference
    const float* w  = (const float*)d_in[3];  // [64,2]
    float* out = (float*)d_out;               // [64,3,32,64]

    dim3 grid(B * CCH);   // 192 workgroups, one (b,c) channel image each
    dim3 block(1024);     // 32 wave32s
    hipLaunchKernelGGL(retina_polar_kernel, grid, block, 0, stream, x, lt, w, out);
}